// GIN_23708219474067
// MI455X (gfx1250) — compile-verified
//
#include <hip/hip_runtime.h>
#include <hip/hip_bf16.h>

typedef __attribute__((ext_vector_type(2))) float v2f;
typedef __attribute__((ext_vector_type(8))) float v8f;

#define BN_EPS 1e-5f

// ---------------------------------------------------------------------------
// BN pass 1: per-feature sum and sum-of-squares of h = x * importance.
// blockDim = 128 (one thread per feature), each block reduces 64 rows in
// registers then does one f32 HW atomic per feature.
// ---------------------------------------------------------------------------
__global__ __launch_bounds__(128) void bn_stats(const float* __restrict__ x,
                                                const float* __restrict__ imp,
                                                float* __restrict__ sums, int N) {
    int f  = threadIdx.x;
    int r0 = blockIdx.x * 64;
    int rmax = N - r0; if (rmax > 64) rmax = 64;
    float s = 0.f, s2 = 0.f;
    for (int r = 0; r < rmax; ++r) {
        int row = r0 + r;
        float v = x[(size_t)row * 128 + f] * imp[row];
        s += v; s2 += v * v;
    }
    unsafeAtomicAdd(&sums[f], s);
    unsafeAtomicAdd(&sums[128 + f], s2);
}

// BN pass 2: fold mean/var/gamma/beta into per-feature scale+shift.
__global__ __launch_bounds__(128) void bn_finalize(const float* __restrict__ sums,
                                                   const float* __restrict__ gamma,
                                                   const float* __restrict__ beta,
                                                   float* __restrict__ sc_sh, float invN) {
    int f = threadIdx.x;
    float mean = sums[f] * invN;
    float var  = sums[128 + f] * invN - mean * mean;
    float sc   = gamma[f] * rsqrtf(var + BN_EPS);
    sc_sh[f]        = sc;
    sc_sh[128 + f]  = beta[f] - mean * sc;
}

// BN pass 3: h0 = (x*imp)*scale + shift, float4 vectorized.
__global__ __launch_bounds__(256) void bn_norm(const float* __restrict__ x,
                                               const float* __restrict__ imp,
                                               const float* __restrict__ sc_sh,
                                               float* __restrict__ h0, long total4) {
    long t = (long)blockIdx.x * blockDim.x + threadIdx.x;
    if (t >= total4) return;
    int row = (int)(t >> 5);          // 32 float4 per row of 128
    int f0  = ((int)t & 31) * 4;
    float im = imp[row];
    float4 v  = *(const float4*)(x + (size_t)row * 128 + f0);
    float4 sc = *(const float4*)(sc_sh + f0);
    float4 sh = *(const float4*)(sc_sh + 128 + f0);
    float4 o;
    o.x = v.x * im * sc.x + sh.x;
    o.y = v.y * im * sc.y + sh.y;
    o.z = v.z * im * sc.z + sh.z;
    o.w = v.w * im * sc.w + sh.w;
    *(float4*)(h0 + (size_t)row * 128 + f0) = o;
}

// ---------------------------------------------------------------------------
// Edge aggregation: agg[dst] += h[src].  32 threads per edge, each thread
// handles 4 contiguous features (float4 gather + 4 HW f32 atomics).
// h and agg both fit in the 192MB L2, so this runs near L2 rate.
// ---------------------------------------------------------------------------
__global__ __launch_bounds__(256) void scatter_add(const int* __restrict__ ei,
                                                   const float* __restrict__ h, int strideH,
                                                   float* __restrict__ agg,
                                                   long E, long total) {
    long gid = (long)blockIdx.x * blockDim.x + threadIdx.x;
    if (gid >= total) return;
    long e   = gid >> 5;
    int lane = (int)(gid & 31);
    int s  = ei[e];
    int dt = ei[E + e];
    float4 hv = *(const float4*)(h + (size_t)s * strideH + lane * 4);
    float* ap = agg + (size_t)dt * 128 + lane * 4;
    unsafeAtomicAdd(ap + 0, hv.x);
    unsafeAtomicAdd(ap + 1, hv.y);
    unsafeAtomicAdd(ap + 2, hv.z);
    unsafeAtomicAdd(ap + 3, hv.w);
}

// ---------------------------------------------------------------------------
// GIN GEMM + bias + tanh via fp32 WMMA (V_WMMA_F32_16X16X4_F32).
// Block = 256 threads = 8 waves; block tile = 16 nodes x 128 out-features,
// each wave owns one 16x16 column tile, K=128 in steps of 4.
// A tile (h+agg) staged in LDS with stride 132 (bank-conflict free).
// ---------------------------------------------------------------------------
__global__ __launch_bounds__(256) void gin_gemm(const float* __restrict__ Hin, int strideIn,
                                                const float* __restrict__ Agg,
                                                const float* __restrict__ W,
                                                const float* __restrict__ bias,
                                                float* __restrict__ Hout, int strideOut) {
    __shared__ float As[16 * 132];
    int node0 = blockIdx.x * 16;

    // Stage 16x128 A tile (h + agg): 512 float4, 2 per thread.
    for (int i = threadIdx.x; i < 512; i += 256) {
        int r  = i >> 5;            // 32 float4 per row
        int c4 = (i & 31) * 4;
        float4 v = *(const float4*)(Hin + (size_t)(node0 + r) * strideIn + c4);
        if (Agg) {
            float4 a = *(const float4*)(Agg + (size_t)(node0 + r) * 128 + c4);
            v.x += a.x; v.y += a.y; v.z += a.z; v.w += a.w;
        }
        *(float4*)(&As[r * 132 + c4]) = v;
    }
    __syncthreads();

    int lane = threadIdx.x & 31;
    int wid  = threadIdx.x >> 5;
    int n0   = wid * 16;
    int row  = lane & 15;             // A: M index
    int koff = (lane >> 4) << 1;      // A/B: K sub-offset (0 or 2)
    int col  = lane & 15;             // B: N index

    v8f acc = {};
    const float* as = &As[row * 132 + koff];
    const float* wp = W + (size_t)koff * 128 + n0 + col;

#pragma unroll
    for (int k0 = 0; k0 < 128; k0 += 4) {
        v2f a; a.x = as[k0];            a.y = as[k0 + 1];
        v2f b; b.x = wp[(size_t)k0 * 128]; b.y = wp[(size_t)(k0 + 1) * 128];
        acc = __builtin_amdgcn_wmma_f32_16x16x4_f32(
            /*neg_a=*/false, a, /*neg_b=*/false, b,
            /*c_mod=*/(short)0, acc, /*reuse_a=*/false, /*reuse_b=*/false);
    }

    // Epilogue: D layout -> VGPR i holds M = i + 8*(lane>=16), N = lane&15.
    int n  = n0 + col;
    float bv = bias ? bias[n] : 0.0f;
    int mbase = (lane >> 4) << 3;
#pragma unroll
    for (int i = 0; i < 8; ++i) {
        int m = mbase + i;
        float v = tanhf(acc[i] + bv);
        Hout[(size_t)(node0 + m) * strideOut + n] = v;
    }
}

// ---------------------------------------------------------------------------
extern "C" void kernel_launch(void* const* d_in, const int* in_sizes, int n_in,
                              void* d_out, int out_size, void* d_ws, size_t ws_size,
                              hipStream_t stream) {
    const float* x     = (const float*)d_in[0];
    const float* imp   = (const float*)d_in[1];
    const int*   ei    = (const int*)d_in[2];
    const float* gamma = (const float*)d_in[3];
    const float* beta  = (const float*)d_in[4];
    const float* Wfc   = (const float*)d_in[5];
    const float* W[5]  = { (const float*)d_in[6],  (const float*)d_in[8],
                           (const float*)d_in[10], (const float*)d_in[12],
                           (const float*)d_in[14] };
    const float* b[5]  = { (const float*)d_in[7],  (const float*)d_in[9],
                           (const float*)d_in[11], (const float*)d_in[13],
                           (const float*)d_in[15] };

    const int  N = in_sizes[0] / 128;
    const long E = in_sizes[2] / 2;
    const int  OUTS = 768;   // 6 concatenated slices of 128

    float* h0    = (float*)d_ws;                    // N*128
    float* agg   = h0 + (size_t)N * 128;            // N*128
    float* sums  = agg + (size_t)N * 128;           // 256
    float* sc_sh = sums + 256;                      // 256
    float* out   = (float*)d_out;

    // ---- BatchNorm (training-mode batch stats) ----
    hipMemsetAsync(sums, 0, 256 * sizeof(float), stream);
    bn_stats<<<(N + 63) / 64, 128, 0, stream>>>(x, imp, sums, N);
    bn_finalize<<<1, 128, 0, stream>>>(sums, gamma, beta, sc_sh, 1.0f / (float)N);
    long total4 = (long)N * 32;
    bn_norm<<<(unsigned)((total4 + 255) / 256), 256, 0, stream>>>(x, imp, sc_sh, h0, total4);

    // ---- 5 GIN layers: aggregate + GEMM+bias+tanh (output slice doubles as
    //      next layer's input) ----
    const float* hin = h0;
    int strideIn = 128;
    for (int l = 0; l < 5; ++l) {
        hipMemsetAsync(agg, 0, (size_t)N * 128 * sizeof(float), stream);
        long tw = E * 32;
        scatter_add<<<(unsigned)((tw + 255) / 256), 256, 0, stream>>>(
            ei, hin, strideIn, agg, E, tw);
        float* hout = out + (size_t)l * 128;
        gin_gemm<<<N / 16, 256, 0, stream>>>(hin, strideIn, agg, W[l], b[l], hout, OUTS);
        hin = hout;
        strideIn = OUTS;
    }

    // ---- Final FC: tanh(h5 @ Wfc), no agg, no bias ----
    gin_gemm<<<N / 16, 256, 0, stream>>>(hin, strideIn, nullptr, Wfc, nullptr,
                                         out + (size_t)5 * 128, OUTS);
}